// GINet_34651796144642
// MI455X (gfx1250) — compile-verified
//
#include <hip/hip_runtime.h>
#include <hip/hip_bf16.h>

typedef __attribute__((ext_vector_type(2))) float v2f;
typedef __attribute__((ext_vector_type(8))) float v8f;

#define FEAT   128
#define K2     256
#define PADN   144   // padded N stride (in 8B pairs): 144*2 % 64 == 32 -> half-waves hit disjoint banks

// ---------------- kernel 0: zero workspace ----------------
__global__ void zero_f32(float* __restrict__ p, size_t n) {
    size_t i = (size_t)blockIdx.x * blockDim.x + threadIdx.x;
    size_t stride = (size_t)gridDim.x * blockDim.x;
    for (; i < n; i += stride) p[i] = 0.0f;
}

// ---------------- kernel 1: edge scatter ----------------
// agg[row] += ea * x[col]   (128 f32, 4 per lane as float4 gather, f32 atomics in L2)
// s[row]   += ea ;  cnt[row] += 1
__global__ void edge_scatter(const float* __restrict__ x,
                             const int*   __restrict__ ei,
                             const float* __restrict__ ea,
                             float* __restrict__ agg,
                             float* __restrict__ s,
                             float* __restrict__ cnt,
                             int E) {
    int lane  = threadIdx.x & 31;
    int wave  = (blockIdx.x * blockDim.x + threadIdx.x) >> 5;
    int nwave = (gridDim.x * blockDim.x) >> 5;
    for (int e = wave; e < E; e += nwave) {
        int   r = ei[e];       // destination (segment id)
        int   c = ei[E + e];   // gathered neighbor
        float w = ea[e];
        const float4 v = ((const float4*)(x + (size_t)c * FEAT))[lane];
        float* dst = agg + (size_t)r * FEAT + lane * 4;
        atomicAdd(dst + 0, w * v.x);
        atomicAdd(dst + 1, w * v.y);
        atomicAdd(dst + 2, w * v.z);
        atomicAdd(dst + 3, w * v.w);
        if (lane == 0) {
            atomicAdd(s + r, w);
            atomicAdd(cnt + r, 1.0f);
        }
    }
}

// ---------------- kernel 2: node GEMM (fp32 WMMA) ----------------
// out[16 rows x 128] per wave:  D = [s*x | agg](16x256) @ W(256x128), then /max(cnt,1) + bias
// LDS holds W interleaved by K-pairs: pair p=k/2 at wp[p*PADN + n] = {W[2p][n], W[2p+1][n]}
// so each B fragment is one conflict-free ds_load_b64 into an even VGPR pair.
__global__ __launch_bounds__(128) void node_gemm(const float* __restrict__ x,
                                                 const float* __restrict__ agg,
                                                 const float* __restrict__ s,
                                                 const float* __restrict__ cnt,
                                                 const float* __restrict__ weight,
                                                 const float* __restrict__ bias,
                                                 float* __restrict__ out,
                                                 int nblocks) {
    extern __shared__ float w_lds[];   // 128 pairs x PADN x 8B = 147,456 B (gfx1250 WGP: 320 KB)

    // cooperative stage of W[256][128] into K-pair-interleaved LDS
    for (int i = threadIdx.x; i < K2 * FEAT; i += blockDim.x) {
        int k = i >> 7, n = i & 127;
        w_lds[(((k >> 1) * PADN + n) << 1) + (k & 1)] = weight[i];
    }
    __syncthreads();

    const v2f* __restrict__ wp = (const v2f*)w_lds;

    const int lane  = threadIdx.x & 31;
    const int wave  = threadIdx.x >> 5;
    const int wgl   = blockIdx.x * (blockDim.x >> 5) + wave;
    const int nwav  = gridDim.x * (blockDim.x >> 5);
    const int m15   = lane & 15;
    const int khi   = lane >> 4;                 // lanes 0-15 -> K {0,1}; lanes 16-31 -> K {2,3}
    const int khalf = khi * 2;
    const int bbase = khi * PADN + m15;          // per-lane base into pair-indexed LDS

    for (int mb = wgl; mb < nblocks; mb += nwav) {
        const int row0  = mb * 16;
        const int myrow = row0 + m15;            // A-fragment row for this lane
        const float sc  = s[myrow];
        const float* __restrict__ xr = x   + (size_t)myrow * FEAT;
        const float* __restrict__ ar = agg + (size_t)myrow * FEAT;

        v8f acc[8] = {};

        // K = 0..127 : A = s[n] * x[n]
        #pragma unroll 2
        for (int k = 0; k < FEAT; k += 4) {
            v2f a;
            a.x = xr[k + khalf + 0] * sc;
            a.y = xr[k + khalf + 1] * sc;
            const int pidx = (k >> 1) * PADN + bbase;
            #pragma unroll
            for (int t = 0; t < 8; t++) {
                v2f b = wp[pidx + t * 16];
                acc[t] = __builtin_amdgcn_wmma_f32_16x16x4_f32(
                    false, a, false, b, (short)0, acc[t], false, false);
            }
        }
        // K = 128..255 : A = agg[n]
        #pragma unroll 2
        for (int k = 0; k < FEAT; k += 4) {
            v2f a;
            a.x = ar[k + khalf + 0];
            a.y = ar[k + khalf + 1];
            const int pidx = (64 + (k >> 1)) * PADN + bbase;
            #pragma unroll
            for (int t = 0; t < 8; t++) {
                v2f b = wp[pidx + t * 16];
                acc[t] = __builtin_amdgcn_wmma_f32_16x16x4_f32(
                    false, a, false, b, (short)0, acc[t], false, false);
            }
        }

        // epilogue: D VGPR v -> (m = v + 8*(lane>=16), n = t*16 + (lane&15))
        const int moff = khi * 8;
        float inv[8];
        #pragma unroll
        for (int v = 0; v < 8; v++)
            inv[v] = 1.0f / fmaxf(cnt[row0 + moff + v], 1.0f);

        #pragma unroll
        for (int t = 0; t < 8; t++) {
            const float bval = bias[t * 16 + m15];
            #pragma unroll
            for (int v = 0; v < 8; v++) {
                const int m = moff + v;
                out[(size_t)(row0 + m) * FEAT + t * 16 + m15] = acc[t][v] * inv[v] + bval;
            }
        }
    }
}

// ---------------- launcher ----------------
extern "C" void kernel_launch(void* const* d_in, const int* in_sizes, int n_in,
                              void* d_out, int out_size, void* d_ws, size_t ws_size,
                              hipStream_t stream) {
    const float* x  = (const float*)d_in[0];
    const int*   ei = (const int*)  d_in[1];
    const float* ea = (const float*)d_in[2];
    const float* w  = (const float*)d_in[3];
    const float* b  = (const float*)d_in[4];
    float* out = (float*)d_out;

    const int N = in_sizes[0] / FEAT;   // 40000
    const int E = in_sizes[2];          // 640000

    float* agg = (float*)d_ws;                      // N * 128
    float* s   = agg + (size_t)N * FEAT;            // N
    float* cnt = s + N;                             // N
    const size_t zn = (size_t)N * FEAT + 2 * (size_t)N;

    zero_f32<<<2048, 256, 0, stream>>>(agg, zn);
    edge_scatter<<<640, 256, 0, stream>>>(x, ei, ea, agg, s, cnt, E);

    const int nblocks = N / 16;                     // 2500
    const int gemm_blocks = (nblocks + 3) / 4;      // 4 waves per block
    node_gemm<<<gemm_blocks, 128, (K2 / 2) * PADN * 2 * sizeof(float), stream>>>(
        x, agg, s, cnt, w, b, out, nblocks);
}